// HGT_28647431864450
// MI455X (gfx1250) — compile-verified
//
#include <hip/hip_runtime.h>
#include <math.h>

#define T_    3
#define R_    4
#define H_    8
#define DK_   16
#define HID_  128
#define INV_SQRT_DK 0.25f
#define SH_REL (R_ * H_ * DK_ * DK_)   // 8192 floats = 32KB

typedef float v2f __attribute__((ext_vector_type(2)));
typedef float v8f __attribute__((ext_vector_type(8)));

// ---- ordered-uint mapping for float atomic max ----
__device__ __forceinline__ unsigned fmap(float f) {
    unsigned u = __float_as_uint(f);
    return (u & 0x80000000u) ? ~u : (u | 0x80000000u);
}
__device__ __forceinline__ float funmap(unsigned u) {
    return (u & 0x80000000u) ? __uint_as_float(u ^ 0x80000000u)
                             : __uint_as_float(~u);
}

// =====================================================================
// Typed linear: Y[n,f] = X[n,:] @ W[type(n),:,f] + b[type(n),f]
// One wave = 16-node x 16-feature tile; V_WMMA_F32_16X16X4_F32 K-loop.
// Three accumulators (one per node type), row-select at the end.
// =====================================================================
__global__ __launch_bounds__(256) void hgt_typed_linear_wmma(
    const float* __restrict__ X, int in_dim,
    const float* __restrict__ W,     // [T, in_dim, HID]
    const float* __restrict__ bias,  // [T, HID]
    const int*   __restrict__ types,
    float*       __restrict__ Y,     // [n, HID]
    int n, int act_tanh)
{
    const int lane  = threadIdx.x & 31;
    const int wid   = threadIdx.x >> 5;
    const int flat  = blockIdx.x * 8 + wid;
    const int ntile = flat >> 3;        // 8 f-tiles per node tile
    const int ftile = flat & 7;
    const int node0 = ntile * 16;
    if (node0 >= n) return;

    const int half = lane >> 4;         // which half-wave
    const int lrow = lane & 15;
    const int fcol = ftile * 16 + lrow;
    int arow = node0 + lrow; if (arow >= n) arow = n - 1;

    // A: lane(<16) row=lrow holds K = k0 + {0,1}; upper half K = k0 + {2,3}
    const float*  xp = X + (size_t)arow * in_dim + (half << 1);
    // B: lane holds column fcol, K rows mirror the A K-split
    const float*  wp = W + (size_t)(half << 1) * HID_ + fcol;
    const size_t  wt = (size_t)in_dim * HID_;   // per-type weight stride

    v8f a0 = {}, a1 = {}, a2 = {};
    for (int k0 = 0; k0 < in_dim; k0 += 4) {
        v2f av; av.x = xp[k0]; av.y = xp[k0 + 1];
        const float* wk = wp + (size_t)k0 * HID_;
        v2f b0; b0.x = wk[0];          b0.y = wk[HID_];
        v2f b1; b1.x = wk[wt];         b1.y = wk[wt + HID_];
        v2f b2; b2.x = wk[2 * wt];     b2.y = wk[2 * wt + HID_];
        a0 = __builtin_amdgcn_wmma_f32_16x16x4_f32(false, av, false, b0, (short)0, a0, false, false);
        a1 = __builtin_amdgcn_wmma_f32_16x16x4_f32(false, av, false, b1, (short)0, a1, false, false);
        a2 = __builtin_amdgcn_wmma_f32_16x16x4_f32(false, av, false, b2, (short)0, a2, false, false);
    }

    // D layout: lane(<16)=col, vgpr r = row r + 8*half
    #pragma unroll
    for (int r = 0; r < 8; ++r) {
        int row = node0 + r + (half << 3);
        if (row >= n) continue;
        int t = types[row];
        float yv = (t == 0) ? a0[r] : (t == 1) ? a1[r] : a2[r];
        yv += bias[t * HID_ + fcol];
        if (act_tanh) yv = tanhf(yv);
        Y[(size_t)row * HID_ + fcol] = yv;
    }
}

// =====================================================================
// Zero/init: agg = 0, mx = mapped(-inf) = 0, denom = 0
// =====================================================================
__global__ __launch_bounds__(256) void hgt_init_kernel(
    float* __restrict__ agg, unsigned* __restrict__ mxu,
    float* __restrict__ denom, int nhid, int nh)
{
    int i = blockIdx.x * blockDim.x + threadIdx.x;
    if (i < nhid) agg[i] = 0.0f;
    if (i < nh)  { mxu[i] = 0u; denom[i] = 0.0f; }
}

// =====================================================================
// Pass 1: logits[e,h] = (q[dst] . (k[src] @ Ratt[r,h])) * pri * 1/sqrt(dk)
//         segment max via ordered-uint atomicMax
// =====================================================================
__global__ __launch_bounds__(256) void hgt_edge_logits(
    const int* __restrict__ src, const int* __restrict__ dst,
    const int* __restrict__ etype,
    const float* __restrict__ kbuf, const float* __restrict__ qbuf,
    const float* __restrict__ Ratt,   // [R,H,DK,DK]  (layer slice)
    const float* __restrict__ pri,    // [R,H]        (layer slice)
    float* __restrict__ ew, unsigned* __restrict__ mxu, int E)
{
    __shared__ float sR[SH_REL];
    __shared__ float sP[R_ * H_];
    for (int i = threadIdx.x; i < SH_REL; i += 256) sR[i] = Ratt[i];
    for (int i = threadIdx.x; i < R_ * H_; i += 256) sP[i] = pri[i];
    __syncthreads();

    int idx = blockIdx.x * 256 + threadIdx.x;
    if (idx >= E * H_) return;
    int e = idx >> 3, h = idx & 7;
    int s = src[e], d = dst[e], r = etype[e];

    const float* kp = kbuf + (size_t)s * HID_ + h * DK_;
    const float* qp = qbuf + (size_t)d * HID_ + h * DK_;
    float kk[DK_], qq[DK_];
    #pragma unroll
    for (int j = 0; j < DK_; j += 4) {
        float4 kv = *(const float4*)(kp + j);
        float4 qv = *(const float4*)(qp + j);
        kk[j] = kv.x; kk[j+1] = kv.y; kk[j+2] = kv.z; kk[j+3] = kv.w;
        qq[j] = qv.x; qq[j+1] = qv.y; qq[j+2] = qv.z; qq[j+3] = qv.w;
    }
    const float* Rh = sR + (r * H_ + h) * (DK_ * DK_);
    float acc = 0.0f;
    #pragma unroll 4
    for (int dd = 0; dd < DK_; ++dd) {
        float t = 0.0f;
        #pragma unroll
        for (int f = 0; f < DK_; ++f) t += Rh[dd * DK_ + f] * qq[f];
        acc += kk[dd] * t;
    }
    acc *= sP[r * H_ + h] * INV_SQRT_DK;
    ew[idx] = acc;
    atomicMax(&mxu[(size_t)d * H_ + h], fmap(acc));
}

// =====================================================================
// Pass 2: a = exp(logit - mx[dst]); segment-sum denom
// =====================================================================
__global__ __launch_bounds__(256) void hgt_edge_exp(
    const int* __restrict__ dst,
    float* __restrict__ ew, const unsigned* __restrict__ mxu,
    float* __restrict__ denom, int E)
{
    int idx = blockIdx.x * 256 + threadIdx.x;
    if (idx >= E * H_) return;
    int e = idx >> 3, h = idx & 7;
    int d = dst[e];
    float m = funmap(mxu[(size_t)d * H_ + h]);
    float a = expf(ew[idx] - m);
    ew[idx] = a;
    atomicAdd(&denom[(size_t)d * H_ + h], a);
}

// =====================================================================
// Pass 3: msg = (a/denom) * (v[src] @ Rmsg[r,h]); scatter-add to agg[dst]
// =====================================================================
__global__ __launch_bounds__(256) void hgt_edge_scatter(
    const int* __restrict__ src, const int* __restrict__ dst,
    const int* __restrict__ etype,
    const float* __restrict__ vbuf,
    const float* __restrict__ Rmsg,   // [R,H,DK,DK] (layer slice)
    const float* __restrict__ ew, const float* __restrict__ denom,
    float* __restrict__ agg, int E)
{
    __shared__ float sM[SH_REL];
    for (int i = threadIdx.x; i < SH_REL; i += 256) sM[i] = Rmsg[i];
    __syncthreads();

    int idx = blockIdx.x * 256 + threadIdx.x;
    if (idx >= E * H_) return;
    int e = idx >> 3, h = idx & 7;
    int s = src[e], d = dst[e], r = etype[e];

    const float* vp = vbuf + (size_t)s * HID_ + h * DK_;
    float vv[DK_];
    #pragma unroll
    for (int j = 0; j < DK_; j += 4) {
        float4 t = *(const float4*)(vp + j);
        vv[j] = t.x; vv[j+1] = t.y; vv[j+2] = t.z; vv[j+3] = t.w;
    }
    float coef = ew[idx] / fmaxf(denom[(size_t)d * H_ + h], 1e-9f);
    const float* Rh = sM + (r * H_ + h) * (DK_ * DK_);
    float vrel[DK_];
    #pragma unroll
    for (int f = 0; f < DK_; ++f) vrel[f] = 0.0f;
    #pragma unroll 4
    for (int dd = 0; dd < DK_; ++dd) {
        float vd = vv[dd];
        #pragma unroll
        for (int f = 0; f < DK_; ++f) vrel[f] += vd * Rh[dd * DK_ + f];
    }
    float* ap = agg + (size_t)d * HID_ + h * DK_;
    #pragma unroll
    for (int f = 0; f < DK_; ++f) atomicAdd(ap + f, coef * vrel[f]);
}

// =====================================================================
// Elementwise exact GELU
// =====================================================================
__global__ __launch_bounds__(256) void hgt_gelu(
    const float* __restrict__ x, float* __restrict__ y, int n)
{
    int i = blockIdx.x * 256 + threadIdx.x;
    if (i >= n) return;
    float v = x[i];
    y[i] = 0.5f * v * (1.0f + erff(v * 0.70710678118654752f));
}

// =====================================================================
// Post: gated skip + optional layernorm + optional relu.
// One wave per node; 4 features per lane; wave32 shfl reductions.
// =====================================================================
__global__ __launch_bounds__(256) void hgt_post(
    const float* __restrict__ trans, const float* __restrict__ hin,
    const int* __restrict__ types, const float* __restrict__ skipL, // [T]
    const float* __restrict__ gamma, const float* __restrict__ beta,
    float* __restrict__ out, int n, int use_norm, int relu)
{
    int wid = threadIdx.x >> 5, lane = threadIdx.x & 31;
    int node = blockIdx.x * 8 + wid;
    if (node >= n) return;
    int t = types[node];
    float alpha = 1.0f / (1.0f + expf(-skipL[t]));
    float o[4];
    #pragma unroll
    for (int j = 0; j < 4; ++j) {
        int f = lane + 32 * j;
        float tr = trans[(size_t)node * HID_ + f];
        float hv = hin[(size_t)node * HID_ + f];
        o[j] = tr * alpha + hv * (1.0f - alpha);
    }
    if (use_norm) {
        float s1 = o[0] + o[1] + o[2] + o[3];
        float s2 = o[0]*o[0] + o[1]*o[1] + o[2]*o[2] + o[3]*o[3];
        #pragma unroll
        for (int off = 16; off > 0; off >>= 1) {
            s1 += __shfl_xor(s1, off, 32);
            s2 += __shfl_xor(s2, off, 32);
        }
        float mu  = s1 * (1.0f / HID_);
        float var = s2 * (1.0f / HID_) - mu * mu;
        float inv = rsqrtf(var + 1e-5f);
        #pragma unroll
        for (int j = 0; j < 4; ++j) {
            int f = lane + 32 * j;
            o[j] = (o[j] - mu) * inv * gamma[t * HID_ + f] + beta[t * HID_ + f];
        }
    }
    #pragma unroll
    for (int j = 0; j < 4; ++j) {
        int f = lane + 32 * j;
        float v = relu ? fmaxf(o[j], 0.0f) : o[j];
        out[(size_t)node * HID_ + f] = v;
    }
}

// =====================================================================
extern "C" void kernel_launch(void* const* d_in, const int* in_sizes, int n_in,
                              void* d_out, int out_size, void* d_ws, size_t ws_size,
                              hipStream_t stream)
{
    (void)n_in; (void)out_size; (void)ws_size;
    const float* x          = (const float*)d_in[0];
    const int*   node_types = (const int*)  d_in[1];
    const int*   edge_index = (const int*)  d_in[2];
    const int*   edge_types = (const int*)  d_in[3];
    const float* W_adapt    = (const float*)d_in[4];
    const float* b_adapt    = (const float*)d_in[5];
    const float* Wk         = (const float*)d_in[6];
    const float* bk         = (const float*)d_in[7];
    const float* Wq         = (const float*)d_in[8];
    const float* bq         = (const float*)d_in[9];
    const float* Wv         = (const float*)d_in[10];
    const float* bv         = (const float*)d_in[11];
    const float* Wa         = (const float*)d_in[12];
    const float* ba         = (const float*)d_in[13];
    const float* rel_pri    = (const float*)d_in[14];
    const float* rel_att    = (const float*)d_in[15];
    const float* rel_msg    = (const float*)d_in[16];
    const float* skip       = (const float*)d_in[17];
    const float* ln_gamma   = (const float*)d_in[18];
    const float* ln_beta    = (const float*)d_in[19];

    const int N  = in_sizes[1];
    const int E  = in_sizes[3];
    const int IN = in_sizes[0] / N;          // 256
    const int* src = edge_index;
    const int* dst = edge_index + E;

    // workspace layout (floats)
    const size_t S  = (size_t)N * HID_;
    const size_t EH = (size_t)E * H_;
    const size_t NH = (size_t)N * H_;
    float* ws    = (float*)d_ws;
    float* hA    = ws;
    float* hB    = ws + S;
    float* kbuf  = ws + 2 * S;               // also reused as gelu(agg)
    float* qbuf  = ws + 3 * S;               // also reused as trans
    float* vbuf  = ws + 4 * S;
    float* agg   = ws + 5 * S;
    float* ew    = ws + 6 * S;
    unsigned* mxu = (unsigned*)(ws + 6 * S + EH);
    float* denom = ws + 6 * S + EH + NH;

    const int tl_blocks   = (N + 15) / 16;               // 8 waves/block = 8 f-tiles
    const int eh_blocks   = (int)((EH + 255) / 256);
    const int nhid_blocks = (int)((S + 255) / 256);
    const int post_blocks = (N + 7) / 8;

    // ---- adapt: h = tanh(typed_linear(x)) ----
    hgt_typed_linear_wmma<<<tl_blocks, 256, 0, stream>>>(
        x, IN, W_adapt, b_adapt, node_types, hA, N, 1);

    for (int l = 0; l < 2; ++l) {
        const float* hin  = l ? hB : hA;
        float*       hout = l ? (float*)d_out : hB;
        const size_t woff = (size_t)l * T_ * HID_ * HID_;
        const size_t boff = (size_t)l * T_ * HID_;
        const float* RattL = rel_att + (size_t)l * SH_REL;
        const float* RmsgL = rel_msg + (size_t)l * SH_REL;
        const float* priL  = rel_pri + (size_t)l * R_ * H_;
        const float* skipL = skip + (size_t)l * T_;

        hgt_typed_linear_wmma<<<tl_blocks, 256, 0, stream>>>(
            hin, HID_, Wk + woff, bk + boff, node_types, kbuf, N, 0);
        hgt_typed_linear_wmma<<<tl_blocks, 256, 0, stream>>>(
            hin, HID_, Wq + woff, bq + boff, node_types, qbuf, N, 0);
        hgt_typed_linear_wmma<<<tl_blocks, 256, 0, stream>>>(
            hin, HID_, Wv + woff, bv + boff, node_types, vbuf, N, 0);

        hgt_init_kernel<<<nhid_blocks, 256, 0, stream>>>(
            agg, mxu, denom, (int)S, (int)NH);

        hgt_edge_logits<<<eh_blocks, 256, 0, stream>>>(
            src, dst, edge_types, kbuf, qbuf, RattL, priL, ew, mxu, E);
        hgt_edge_exp<<<eh_blocks, 256, 0, stream>>>(
            dst, ew, mxu, denom, E);
        hgt_edge_scatter<<<eh_blocks, 256, 0, stream>>>(
            src, dst, edge_types, vbuf, RmsgL, ew, denom, agg, E);

        // gelu(agg) -> kbuf (k no longer needed)
        hgt_gelu<<<nhid_blocks, 256, 0, stream>>>(agg, kbuf, (int)S);
        // trans = typed_linear(gelu(agg)) -> qbuf
        hgt_typed_linear_wmma<<<tl_blocks, 256, 0, stream>>>(
            kbuf, HID_, Wa + woff, ba + boff, node_types, qbuf, N, 0);
        // skip mix + (layer0: layernorm+relu)
        hgt_post<<<post_blocks, 256, 0, stream>>>(
            qbuf, hin, node_types, skipL, ln_gamma, ln_beta,
            hout, N, l == 0 ? 1 : 0, l == 0 ? 1 : 0);
    }
}